// AssociativeScanGateLoop_68272800137227
// MI455X (gfx1250) — compile-verified
//
#include <hip/hip_runtime.h>
#include <cstdint>
#include <cstddef>

#define DEV __device__ __forceinline__

// Problem constants (fixed by the reference): x is (4, 8192, 3*512) fp32.
constexpr int B_   = 4;
constexpr int T_   = 8192;
constexpr int D_   = 512;
constexpr int ROW  = 3 * D_;          // 1536 floats per (b,t) row
constexpr int TC   = 64;              // timesteps per chunk
constexpr int NCH  = T_ / TC;         // 128 chunks
constexpr int TSUB = 4;               // timesteps per LDS tile (pass 3)
constexpr int NTILES = TC / TSUB;     // 16 tiles per chunk
constexpr int THREADS = 128;          // 128 threads * 4 channels = 512 = D_
constexpr int UNITS_PER_TILE = (TSUB * ROW) / 4;       // 1536 float4 units
constexpr int LOADS_PER_THREAD = UNITS_PER_TILE / THREADS; // 12

// Native clang vector types (the builtins demand real vectors, not HIP float4).
typedef int   v4i __attribute__((vector_size(4 * sizeof(int))));
typedef float v4f __attribute__((vector_size(4 * sizeof(float))));
typedef __attribute__((address_space(1))) v4i* as1_v4ip;
typedef __attribute__((address_space(3))) v4i* as3_v4ip;
typedef __attribute__((address_space(3))) char* as3_cp;

// ---------------- fast transcendentals (CDNA5 v_tanh_f32 when available) ---
DEV float fast_tanh(float x) {
#if __has_builtin(__builtin_amdgcn_tanhf)
  return __builtin_amdgcn_tanhf(x);
#elif __has_builtin(__builtin_amdgcn_tanh_f32)
  return __builtin_amdgcn_tanh_f32(x);
#else
  float cx = fminf(fmaxf(x, -15.0f), 15.0f);
  float e  = __expf(2.0f * cx);
  return (e - 1.0f) * __builtin_amdgcn_rcpf(e + 1.0f);
#endif
}
DEV float fast_sigmoid(float x) {
  // sigmoid(x) = 0.5*tanh(x/2) + 0.5  -> one trans op + one FMA, no division
  return __builtin_fmaf(0.5f, fast_tanh(0.5f * x), 0.5f);
}

// ---------------- small float4 helpers --------------------------------------
struct f4 { float v[4]; };
DEV f4 load4(const float* p) {
  v4f t = *reinterpret_cast<const v4f*>(p);
  f4 r; r.v[0] = t[0]; r.v[1] = t[1]; r.v[2] = t[2]; r.v[3] = t[3]; return r;
}
DEV f4 load4_nt(const float* p) {   // non-temporal (streaming) load
  v4f t = __builtin_nontemporal_load(reinterpret_cast<const v4f*>(p));
  f4 r; r.v[0] = t[0]; r.v[1] = t[1]; r.v[2] = t[2]; r.v[3] = t[3]; return r;
}
DEV void store4(float* p, const f4& a) {
  v4f t = {a.v[0], a.v[1], a.v[2], a.v[3]};
  *reinterpret_cast<v4f*>(p) = t;
}
DEV void store4_nt(float* p, const f4& a) {  // non-temporal (streaming) store
  v4f t = {a.v[0], a.v[1], a.v[2], a.v[3]};
  __builtin_nontemporal_store(t, reinterpret_cast<v4f*>(p));
}

// ---------------- CDNA5 async global->LDS copy (ASYNCcnt tracked) ----------
DEV void async_copy_b128(float* lds_dst, const float* g_src) {
#if __has_builtin(__builtin_amdgcn_global_load_async_to_lds_b128)
  __builtin_amdgcn_global_load_async_to_lds_b128(
      (as1_v4ip)g_src, (as3_v4ip)lds_dst, 0, 0);
#else
  unsigned lds_off = (unsigned)(unsigned long long)(as3_cp)lds_dst;
  unsigned long long ga = (unsigned long long)g_src;
  asm volatile("global_load_async_to_lds_b128 %0, %1, off"
               :: "v"(lds_off), "v"(ga) : "memory");
#endif
}

#if __has_builtin(__builtin_amdgcn_s_wait_asynccnt)
#define WAIT_ASYNC(n) __builtin_amdgcn_s_wait_asynccnt(n)
#else
#define WAIT_ASYNC(n) asm volatile("s_wait_asynccnt %0" :: "i"(n) : "memory")
#endif

// ============================================================================
// Pass 3 (emitted first so the disasm snippet shows the async pipeline):
// re-scan each chunk from its carry and emit tanh(h)*sigmoid(o).
// Rows x[b,t,:] are contiguous 6KB and the block covers all 512 channels, so
// tiles of TSUB rows are contiguous 24KB blocks streamed into LDS with
// double-buffered global_load_async_to_lds_b128 (ASYNCcnt).
// ============================================================================
__global__ __launch_bounds__(THREADS)
void lin_scan_pass3(const float* __restrict__ x, const float* __restrict__ Carry,
                    float* __restrict__ out) {
  const int c   = blockIdx.x;
  const int b   = blockIdx.y;
  const int tid = threadIdx.x;
  const int d   = tid * 4;

  const float* gbase = x   + ((size_t)b * T_ + (size_t)c * TC) * ROW;
  float*       obase = out + ((size_t)b * T_ + (size_t)c * TC) * D_;

  __shared__ __align__(16) float buf[2][TSUB * ROW];   // 2 * 24KB = 48KB LDS

  // Prologue: async-load tile 0 into buf[0].
#pragma unroll
  for (int k = 0; k < LOADS_PER_THREAD; ++k) {
    const int unit = tid + k * THREADS;
    async_copy_b128(&buf[0][unit * 4], gbase + (size_t)unit * 4);
  }

  f4 h = load4(Carry + ((size_t)b * NCH + c) * D_ + d);

  for (int tile = 0; tile < NTILES; ++tile) {
    const int cur = tile & 1;
    if (tile + 1 < NTILES) {
      // Prefetch next tile into the other buffer while this one is consumed.
      const float* src = gbase + (size_t)(tile + 1) * TSUB * ROW;
#pragma unroll
      for (int k = 0; k < LOADS_PER_THREAD; ++k) {
        const int unit = tid + k * THREADS;
        async_copy_b128(&buf[cur ^ 1][unit * 4], src + (size_t)unit * 4);
      }
      WAIT_ASYNC(LOADS_PER_THREAD);   // only the 12 just-issued may remain
    } else {
      WAIT_ASYNC(0);
    }
    __syncthreads();                  // tile `tile` fully resident in LDS

#pragma unroll
    for (int tt = 0; tt < TSUB; ++tt) {
      const float* row = &buf[cur][tt * ROW];
      f4 u  = load4(row + d);            // ds_load_b128
      f4 g  = load4(row + D_ + d);
      f4 og = load4(row + 2 * D_ + d);
      f4 o;
#pragma unroll
      for (int k = 0; k < 4; ++k) {
        float ig = fast_sigmoid(g.v[k]);
        float fg = 1.0f - ig;
        h.v[k] = __builtin_fmaf(fg, h.v[k], ig * fast_tanh(u.v[k]));
        o.v[k] = fast_tanh(h.v[k]) * fast_sigmoid(og.v[k]);
      }
      // Output is write-once / zero-reuse: non-temporal store keeps the 67MB
      // stream from displacing x lines in the 192MB L2.
      store4_nt(obase + (size_t)(tile * TSUB + tt) * D_ + d, o);
    }
    __syncthreads();                  // buffer free before it is refilled
  }
}

// ============================================================================
// Pass 1: per-chunk scan -> chunk aggregates (A = prod f, H = chunk-local h).
// Reads only u and input-gate thirds of x (2/3 of the bytes), non-temporally
// (reuse distance of this data is the whole 134MB pass > 192MB L2 anyway).
// ============================================================================
__global__ __launch_bounds__(THREADS)
void lin_scan_pass1(const float* __restrict__ x,
                    float* __restrict__ Agg, float* __restrict__ Hgg) {
  const int c = blockIdx.x;           // chunk
  const int b = blockIdx.y;           // batch
  const int d = threadIdx.x * 4;      // 4 consecutive channels per thread
  const float* base = x + ((size_t)b * T_ + (size_t)c * TC) * ROW;

  f4 A, H;
#pragma unroll
  for (int k = 0; k < 4; ++k) { A.v[k] = 1.0f; H.v[k] = 0.0f; }

#pragma unroll 4
  for (int t = 0; t < TC; ++t) {
    const float* row = base + (size_t)t * ROW;
    f4 u = load4_nt(row + d);         // pre-activation input
    f4 g = load4_nt(row + D_ + d);    // input-gate logits
#pragma unroll
    for (int k = 0; k < 4; ++k) {
      float ig = fast_sigmoid(g.v[k]);
      float fg = 1.0f - ig;
      H.v[k] = __builtin_fmaf(fg, H.v[k], ig * fast_tanh(u.v[k]));
      A.v[k] *= fg;
    }
  }
  const size_t idx = ((size_t)b * NCH + c) * D_ + d;
  store4(Agg + idx, A);
  store4(Hgg + idx, H);
}

// ============================================================================
// Pass 2: exclusive scan of chunk aggregates along chunks -> per-chunk carry.
// Tiny: B blocks * 128 threads, NCH iterations.
// ============================================================================
__global__ __launch_bounds__(THREADS)
void lin_scan_pass2(const float* __restrict__ Agg, const float* __restrict__ Hgg,
                    float* __restrict__ Carry) {
  const int b = blockIdx.x;
  const int d = threadIdx.x * 4;
  f4 carry;
#pragma unroll
  for (int k = 0; k < 4; ++k) carry.v[k] = 0.0f;
  for (int c = 0; c < NCH; ++c) {
    const size_t idx = ((size_t)b * NCH + c) * D_ + d;
    f4 a = load4(Agg + idx);
    f4 h = load4(Hgg + idx);
    store4(Carry + idx, carry);       // exclusive: carry INTO chunk c
#pragma unroll
    for (int k = 0; k < 4; ++k)
      carry.v[k] = __builtin_fmaf(a.v[k], carry.v[k], h.v[k]);
  }
}

// ============================================================================
extern "C" void kernel_launch(void* const* d_in, const int* in_sizes, int n_in,
                              void* d_out, int out_size, void* d_ws, size_t ws_size,
                              hipStream_t stream) {
  (void)in_sizes; (void)n_in; (void)out_size; (void)ws_size;

  const float* x   = (const float*)d_in[0];
  float*       out = (float*)d_out;

  // Workspace layout: A | H | Carry, each B*NCH*D floats (1MB each, 3MB total).
  const size_t n = (size_t)B_ * NCH * D_;
  float* Agg   = (float*)d_ws;
  float* Hgg   = Agg + n;
  float* Carry = Hgg + n;

  dim3 grid(NCH, B_);
  lin_scan_pass1<<<grid, THREADS, 0, stream>>>(x, Agg, Hgg);
  lin_scan_pass2<<<dim3(B_), THREADS, 0, stream>>>(Agg, Hgg, Carry);
  lin_scan_pass3<<<grid, THREADS, 0, stream>>>(x, Carry, out);
}